// GroupBasis_12352325943493
// MI455X (gfx1250) — compile-verified
//
#include <hip/hip_runtime.h>
#include <hip/hip_bf16.h>

typedef __attribute__((ext_vector_type(2))) float v2f;
typedef __attribute__((ext_vector_type(8))) float v8f;

#define BS_   32
#define N_    4096
#define D_    16
#define K_    8
#define NPTS  (BS_ * N_)
#define WPB   8          // waves per block (256 threads)

// ---------------------------------------------------------------------------
// Kernel 1: normalize the Lie basis.
// factor_k = sqrt(|sum B_k.B_k| / D) + 1e-6 ; nb_k = B_k / factor_k
// ---------------------------------------------------------------------------
__global__ void normalize_basis_kernel(const float* __restrict__ basis,
                                       float* __restrict__ nb) {
    __shared__ float rfac[K_];
    int t = threadIdx.x;
    if (t < K_) {
        const float* bk = basis + t * D_ * D_;
        float s = 0.f;
        for (int i = 0; i < D_ * D_; ++i) { float v = bk[i]; s += v * v; }
        rfac[t] = 1.0f / (sqrtf(fabsf(s) / (float)D_) + 1e-6f);
    }
    __syncthreads();
    for (int i = t; i < K_ * D_ * D_; i += 64) {
        nb[i] = basis[i] * rfac[i >> 8];   // i/(D*D)
    }
}

// ---------------------------------------------------------------------------
// XOR-16 half-wave exchange on the VALU (v_permlanex16_b32, identity selects).
// ---------------------------------------------------------------------------
__device__ __forceinline__ float xor16(float v) {
#if __has_builtin(__builtin_amdgcn_permlanex16)
    int i = __float_as_int(v);
    i = __builtin_amdgcn_permlanex16(i, i, 0x76543210, 0xFEDCBA98, true, false);
    return __int_as_float(i);
#else
    return __shfl_xor(v, 16, 32);
#endif
}

// ---------------------------------------------------------------------------
// D-layout (WMMA C/D, 8 VGPRs) -> B-fragments (4x16, 2 VGPRs per K-tile).
// D layout: lane l (n=l&15, h=l>>4) holds element (M = r + 8h, N = n) in reg r.
// Sender-side pre-select => 4 exchanges + 12 cndmasks.
// ---------------------------------------------------------------------------
__device__ __forceinline__ void d_to_b(const float T[8], int h,
                                       v2f& b0, v2f& b1, v2f& b2, v2f& b3) {
    float rc0 = xor16(h ? T[0] : T[2]);
    float rc1 = xor16(h ? T[1] : T[3]);
    float rc2 = xor16(h ? T[4] : T[6]);
    float rc3 = xor16(h ? T[5] : T[7]);
    b0[0] = h ? rc0  : T[0];  b0[1] = h ? rc1  : T[1];   // K rows {0,1}|{2,3}
    b1[0] = h ? rc2  : T[4];  b1[1] = h ? rc3  : T[5];   // K rows {4,5}|{6,7}
    b2[0] = h ? T[2] : rc0;   b2[1] = h ? T[3] : rc1;    // K rows {8,9}|{10,11}
    b3[0] = h ? T[6] : rc2;   b3[1] = h ? T[7] : rc3;    // K rows {12,13}|{14,15}
}

// 16x16 f32 matmul: D = A*B + Cinit via 4 chained k=4 WMMAs
__device__ __forceinline__ v8f mm16c(const v2f a[4], v2f b0, v2f b1, v2f b2,
                                     v2f b3, v8f acc) {
    acc = __builtin_amdgcn_wmma_f32_16x16x4_f32(false, a[0], false, b0,
                                                (short)0, acc, false, false);
    acc = __builtin_amdgcn_wmma_f32_16x16x4_f32(false, a[1], false, b1,
                                                (short)0, acc, false, false);
    acc = __builtin_amdgcn_wmma_f32_16x16x4_f32(false, a[2], false, b2,
                                                (short)0, acc, false, false);
    acc = __builtin_amdgcn_wmma_f32_16x16x4_f32(false, a[3], false, b3,
                                                (short)0, acc, false, false);
    return acc;
}

// LDS round-trip: matrix in D layout -> A-fragments (the true transpose path)
__device__ __forceinline__ void d_to_a_lds(float* m, const float Td[8],
                                           int n, int h, v2f aq[4]) {
#pragma unroll
    for (int r = 0; r < 8; ++r)
        m[(r + 8 * h) * 17 + n] = Td[r];
    asm volatile("s_wait_dscnt 0x0" ::: "memory");   // intra-wave LDS dep
#pragma unroll
    for (int t = 0; t < 4; ++t) {
#pragma unroll
        for (int v = 0; v < 2; ++v)
            aq[t][v] = m[n * 17 + 4 * t + 2 * h + v];
    }
    asm volatile("s_wait_dscnt 0x0" ::: "memory");   // loads before tile reuse
}

// ---------------------------------------------------------------------------
// Kernel 2: one wave32 per (b, n) point.
//   S  = sum_k c_k * nb_k ; Sc = S/4
//   exp(Sc) ~= A0 + S3*A1 + S3^2*A2 + c9*S3^3   (Paterson-Stockmeyer, deg 9)
//     S2 = Sc^2, S3 = Sc^3
//     A0 = I + Sc + S2/2
//     A1 = I/6 + Sc/24 + S2/120
//     A2 = I/720 + Sc/5040 + S2/40320,  c9 = 1/362880
//   Horner (all terms commute; leading term is scalar*I so it is VALU-only):
//     H = c9*S3 + A2 ; H = S3*H + A1 ; H = S3*H + A0   (2 matmuls)
//   exp(S) = (exp(Sc))^4 : two squarings.
//   Total: 6 matmuls = 24 v_wmma_f32_16x16x4_f32 per point.
// ---------------------------------------------------------------------------
__global__ __launch_bounds__(256) void expm_apply_kernel(
    const float* __restrict__ x, const float* __restrict__ y,
    const float* __restrict__ coeffs, const float* __restrict__ nb,
    float* __restrict__ out) {
    __shared__ float nb_lds[K_ * 272];   // row-major:  k*272 + row*17 + col
    __shared__ float nbT_lds[K_ * 272];  // transposed: k*272 + col*17 + row
    __shared__ float mat[WPB][16 * 17];  // per-wave transpose tile
    __shared__ float vecb[WPB][32];      // per-wave x|y staging

    const int tid = threadIdx.x;
    for (int i = tid; i < K_ * D_ * D_; i += 256) {
        int k = i >> 8, rem = i & 255;
        int row = rem >> 4, col = rem & 15;
        float v = nb[i];
        nb_lds [k * 272 + row * 17 + col] = v;
        nbT_lds[k * 272 + col * 17 + row] = v;
    }
    __syncthreads();

    const int wave = tid >> 5;
    const int lane = tid & 31;
    const int n    = lane & 15;     // column (B/C/D layouts), row (A layout)
    const int h    = lane >> 4;
    const int p    = blockIdx.x * WPB + wave;      // point index, always < NPTS

    // stage x|y early (lanes 0-15: x, 16-31: y)
    const float* src = h ? (y + p * D_) : (x + p * D_);
    vecb[wave][lane] = src[n];

    // coefficients pre-scaled by 1/4 (scaling step)
    float c[K_];
    const float* cp = coeffs + p * K_;
#pragma unroll
    for (int k = 0; k < K_; ++k) c[k] = cp[k] * 0.25f;

    // Build Sc directly in two layouts from the LDS-cached basis:
    //   A-fragments (row-major reads): a_sc[t][v] = Sc(row=n, col=4t+2h+v)
    //   D layout (transposed-copy reads, contiguous in r => wide LDS loads):
    //     scd[r] = Sc(row=r+8h, col=n) = sum_k c_k * nbT[k][n][r+8h]
    v2f a_sc[4];
#pragma unroll
    for (int t = 0; t < 4; ++t) {
#pragma unroll
        for (int v = 0; v < 2; ++v) {
            int col = 4 * t + 2 * h + v;
            float s = 0.f;
#pragma unroll
            for (int k = 0; k < K_; ++k)
                s += c[k] * nb_lds[k * 272 + n * 17 + col];
            a_sc[t][v] = s;
        }
    }
    float scd[8];
#pragma unroll
    for (int r = 0; r < 8; ++r) scd[r] = 0.f;
#pragma unroll
    for (int k = 0; k < K_; ++k) {
        const float* rowp = &nbT_lds[k * 272 + n * 17 + 8 * h];
#pragma unroll
        for (int r = 0; r < 8; ++r)
            scd[r] += c[k] * rowp[r];
    }

    v2f b0, b1, b2, b3;

    // S2 = Sc * Sc
    d_to_b(scd, h, b0, b1, b2, b3);
    v8f z = {0.f, 0.f, 0.f, 0.f, 0.f, 0.f, 0.f, 0.f};
    v8f acc = mm16c(a_sc, b0, b1, b2, b3, z);
    float s2d[8];
#pragma unroll
    for (int r = 0; r < 8; ++r) s2d[r] = acc[r];

    // S3 = Sc * S2
    d_to_b(s2d, h, b0, b1, b2, b3);
    acc = mm16c(a_sc, b0, b1, b2, b3, z);
    float s3d[8];
#pragma unroll
    for (int r = 0; r < 8; ++r) s3d[r] = acc[r];

    // S3 A-fragments via LDS transpose (fixed A side for all Horner steps)
    float* m = mat[wave];
    v2f a_s3[4];
    d_to_a_lds(m, s3d, n, h, a_s3);

    // A0, A1 (v8f: used as WMMA C operands), A2 folded into H immediately
    float H[8];
    v8f A0v, A1v;
#pragma unroll
    for (int r = 0; r < 8; ++r) {
        float id = ((r + 8 * h) == n) ? 1.0f : 0.0f;
        A0v[r] = id + scd[r] + 0.5f * s2d[r];
        A1v[r] = (1.f / 6.f) * id + (1.f / 24.f) * scd[r] + (1.f / 120.f) * s2d[r];
        float A2 = (1.f / 720.f) * id + (1.f / 5040.f) * scd[r] +
                   (1.f / 40320.f) * s2d[r];
        H[r] = (1.f / 362880.f) * s3d[r] + A2;          // H = c9*S3 + A2
    }

    // H = S3*H + A1 ; H = S3*H + A0   (the +A rides in the WMMA C operand)
    d_to_b(H, h, b0, b1, b2, b3);
    acc = mm16c(a_s3, b0, b1, b2, b3, A1v);
#pragma unroll
    for (int r = 0; r < 8; ++r) H[r] = acc[r];

    d_to_b(H, h, b0, b1, b2, b3);
    acc = mm16c(a_s3, b0, b1, b2, b3, A0v);
    float R[8];
#pragma unroll
    for (int r = 0; r < 8; ++r) R[r] = acc[r];          // R ~= exp(S/4)

    // square twice: exp(S) = R^4
#pragma unroll
    for (int sq = 0; sq < 2; ++sq) {
        v2f aq[4];
        d_to_a_lds(m, R, n, h, aq);
        d_to_b(R, h, b0, b1, b2, b3);
        acc = mm16c(aq, b0, b1, b2, b3, z);
#pragma unroll
        for (int r = 0; r < 8; ++r) R[r] = acc[r];
    }

    // apply E to x (lanes 0-15) and y (lanes 16-31)
#pragma unroll
    for (int r = 0; r < 8; ++r)
        m[(r + 8 * h) * 17 + n] = R[r];
    asm volatile("s_wait_dscnt 0x0" ::: "memory");

    float s = 0.f;
#pragma unroll
    for (int j = 0; j < 16; ++j)
        s += m[n * 17 + j] * vecb[wave][h * 16 + j];

    out[h * (NPTS * D_) + p * D_ + n] = s;
}

// ---------------------------------------------------------------------------
extern "C" void kernel_launch(void* const* d_in, const int* in_sizes, int n_in,
                              void* d_out, int out_size, void* d_ws, size_t ws_size,
                              hipStream_t stream) {
    const float* x      = (const float*)d_in[0];
    const float* y      = (const float*)d_in[1];
    const float* coeffs = (const float*)d_in[2];
    const float* basis  = (const float*)d_in[3];
    float* nb  = (float*)d_ws;                 // 2048 floats of scratch
    float* out = (float*)d_out;

    normalize_basis_kernel<<<1, 64, 0, stream>>>(basis, nb);
    expm_apply_kernel<<<NPTS / WPB, 256, 0, stream>>>(x, y, coeffs, nb, out);
}